// SatPyramid_44160853737721
// MI455X (gfx1250) — compile-verified
//
#include <hip/hip_runtime.h>

typedef __attribute__((ext_vector_type(2))) float v2f;
typedef __attribute__((ext_vector_type(4))) float v4f;
typedef __attribute__((ext_vector_type(8))) float v8f;
typedef __attribute__((ext_vector_type(4))) unsigned int u32x4;
typedef __attribute__((ext_vector_type(4))) int i32x4;
typedef __attribute__((ext_vector_type(8))) int i32x8;

#if __has_builtin(__builtin_amdgcn_tensor_load_to_lds) && __has_builtin(__builtin_amdgcn_s_wait_tensorcnt)
#define HAVE_TDM 1
#else
#define HAVE_TDM 0
#endif

static constexpr int IMG_H   = 512;
static constexpr int IMG_W   = 512;
static constexpr int TILE    = 64;
static constexpr int RAD     = 7;     // (15-1)/2
static constexpr int LROWS   = 80;    // 78 rows used + 2 zero pad
static constexpr int LSTRIDE = 84;    // bank-conflict-avoiding LDS row pitch (bytes: 336 = 21*16)

// One pooling iteration: src -> threshold -> 15x15 box avg -> dstFeat, outAcc += result.
__global__ __launch_bounds__(256) void satpyr_iter(
    const float* __restrict__ src,
    float* __restrict__ dstFeat,
    float* __restrict__ outAcc,
    int accumulate)
{
    __shared__ __attribute__((aligned(16))) float ldsIn[LROWS * LSTRIDE]; // thresholded input tile + halo
    __shared__ __attribute__((aligned(16))) float ldsH [LROWS * LSTRIDE]; // horizontal box-sum intermediate

    const int plane = blockIdx.z;
    const int i0 = blockIdx.y * TILE;         // tile origin (output space)
    const int j0 = blockIdx.x * TILE;
    const float* __restrict__ splane = src     + (size_t)plane * IMG_H * IMG_W;
    float*       __restrict__ fplane = dstFeat + (size_t)plane * IMG_H * IMG_W;
    float*       __restrict__ oplane = outAcc  + (size_t)plane * IMG_H * IMG_W;

    const int tid = threadIdx.x;

    // Interior test: TDM path reads rows [i0-7, i0+70] and cols [j0-7, j0+76] raw.
    const bool interior = (i0 >= RAD) && (i0 + 71 <= IMG_H) &&
                          (j0 >= RAD) && (j0 + LSTRIDE - RAD <= IMG_W);

#if HAVE_TDM
    if (interior) {
        // Zero the two pad rows (disjoint from the TDM-written rows 0..77 -> no race).
        for (int idx = tid; idx < 2 * LSTRIDE; idx += 256)
            ldsIn[78 * LSTRIDE + idx] = 0.0f;

        if (tid < 32) {
            // ---- Tensor DMA: one instruction moves the 84x78 fp32 tile into LDS ----
            // tile_dim0 = 84 == LDS row pitch, so rows land on stride-84 rows directly.
            const uint64_t ga = (uint64_t)(uintptr_t)(splane + (size_t)(i0 - RAD) * IMG_W + (j0 - RAD));
            const uint32_t ldsAddr = (uint32_t)(uintptr_t)&ldsIn[0];

            u32x4 g0;
            g0[0] = 1u;                                              // count=1, user D#
            g0[1] = ldsAddr;                                         // lds_addr (bytes)
            g0[2] = (uint32_t)ga;                                    // global_addr[31:0]
            g0[3] = (uint32_t)((ga >> 32) & 0x01FFFFFFu) | (2u << 30); // global_addr[56:32] | type=2

            const uint32_t dim0 = IMG_W, dim1 = IMG_H;               // tensor extents (floats)
            const uint32_t str0 = IMG_W;                             // row stride (floats)
            const uint32_t str1 = IMG_W * IMG_H;                     // plane stride (unused, 2D)
            i32x8 g1;
            g1[0] = (int)(2u << 16);                                 // data_size=4B; mask=0; no pad/iter
            g1[1] = (int)((dim0 & 0xFFFFu) << 16);                   // tensor_dim0[15:0]
            g1[2] = (int)(((dim0 >> 16) & 0xFFFFu) | ((dim1 & 0xFFFFu) << 16));
            g1[3] = (int)(((dim1 >> 16) & 0xFFFFu) | (84u << 16));   // tile_dim0 = 84
            g1[4] = (int)78u;                                        // tile_dim1 = 78, tile_dim2 = 0
            g1[5] = (int)str0;                                       // tensor_dim0_stride[31:0]
            g1[6] = (int)((str1 & 0xFFFFu) << 16);                   // stride0 hi=0 | stride1[15:0]
            g1[7] = (int)(str1 >> 16);                               // stride1[47:16]

            i32x4 gz4 = {};                                          // groups 2/3 unused (2D tensor)
            i32x8 gz8 = {};                                          // extra group (6-arg toolchain form)
            __builtin_amdgcn_tensor_load_to_lds(g0, g1, gz4, gz4, gz8, 0);
            __builtin_amdgcn_s_wait_tensorcnt(0);                    // issuing wave's TENSORcnt
        }
        __syncthreads();

        // In-place threshold over the tile (vectorized; zeros stay zero).
        v4f* p = (v4f*)ldsIn;
        for (int idx = tid; idx < (LROWS * LSTRIDE) / 4; idx += 256) {
            v4f v = p[idx];
            v.x = (v.x > 0.7f) ? v.x : 0.0f;
            v.y = (v.y > 0.7f) ? v.y : 0.0f;
            v.z = (v.z > 0.7f) ? v.z : 0.0f;
            v.w = (v.w > 0.7f) ? v.w : 0.0f;
            p[idx] = v;
        }
    } else
#endif
    {
        // ---- Boundary (or no-TDM) path: scalar load + zero-pad + threshold ----
        for (int idx = tid; idx < 80 * 80; idx += 256) {
            const int li = idx / 80;
            const int lj = idx - li * 80;
            const int gi = i0 - RAD + li;
            const int gj = j0 - RAD + lj;
            float v = 0.0f;
            if (li < 78 && lj < 78 && gi >= 0 && gi < IMG_H && gj >= 0 && gj < IMG_W) {
                const float f = splane[gi * IMG_W + gj];
                v = (f > 0.7f) ? f : 0.0f;
            }
            ldsIn[li * LSTRIDE + lj] = v;
        }
    }
    __syncthreads();

    const int wave  = tid >> 5;
    const int lane  = tid & 31;
    const int n     = lane & 15;   // N (B/C/D col, and A row index)
    const int khalf = lane >> 4;   // which K-half / M-half this lane holds

    // ---- Horizontal pass: Htmp[li][cj] = sum_{t=0..14} in[li][cj+t] -------
    // 5 row-groups x 4 col-groups of 16x16 subtiles; D = sum_c A_c(16x4) x Band_c(4x16)
    for (int t = wave; t < 20; t += 8) {
        const int rg = t >> 2;                 // local rows 16rg..16rg+15
        const int cg = t & 3;                  // output cols 16cg..16cg+15
        v8f acc = {};
        const float* abase = &ldsIn[(rg * 16 + n) * LSTRIDE + cg * 16];
        #pragma unroll
        for (int c = 0; c < 8; ++c) {
            const int k0 = 4 * c + 2 * khalf;      // K index of pair element 0
            const int k1 = k0 + 1;                 // K index of pair element 1
            v2f a = *(const v2f*)(abase + k0);     // in[row][cg*16 + k0 .. k0+1]
            v2f b;
            b.x = (k0 >= n && k0 <= n + 14) ? 1.0f : 0.0f;  // constant band matrix
            b.y = (k1 >= n && k1 <= n + 14) ? 1.0f : 0.0f;
            acc = __builtin_amdgcn_wmma_f32_16x16x4_f32(
                false, a, false, b, (short)0, acc, false, false);
        }
        // D layout: element e, this lane -> M = e + 8*khalf, N = n
        float* hbase = &ldsH[(rg * 16 + 8 * khalf) * LSTRIDE + cg * 16 + n];
        #pragma unroll
        for (int e = 0; e < 8; ++e)
            hbase[e * LSTRIDE] = acc[e];
    }
    __syncthreads();

    // ---- Vertical pass: out[oi][cj] = (1/225) * sum_{t=0..14} Htmp[oi+t][cj] ----
    // 4x4 16x16 subtiles; D = sum_c Band_c(16x4) x Htmp_c(4x16)
    const float inv = 1.0f / 225.0f;
    for (int t = wave; t < 16; t += 8) {
        const int rg = t >> 2;
        const int cg = t & 3;
        v8f acc = {};
        #pragma unroll
        for (int c = 0; c < 8; ++c) {
            const int k0 = 4 * c + 2 * khalf;
            const int k1 = k0 + 1;
            v2f a;                                           // band matrix rows (M = n)
            a.x = (k0 >= n && k0 <= n + 14) ? 1.0f : 0.0f;
            a.y = (k1 >= n && k1 <= n + 14) ? 1.0f : 0.0f;
            v2f b;                                           // Htmp rows k0,k1 col n
            b.x = ldsH[(rg * 16 + k0) * LSTRIDE + cg * 16 + n];
            b.y = ldsH[(rg * 16 + k1) * LSTRIDE + cg * 16 + n];
            acc = __builtin_amdgcn_wmma_f32_16x16x4_f32(
                false, a, false, b, (short)0, acc, false, false);
        }
        const int gi0 = i0 + rg * 16 + 8 * khalf;
        const int gj  = j0 + cg * 16 + n;
        #pragma unroll
        for (int e = 0; e < 8; ++e) {
            const float f = acc[e] * inv;
            const size_t off = (size_t)(gi0 + e) * IMG_W + gj;
            fplane[off] = f;
            oplane[off] = accumulate ? (oplane[off] + f) : f;
        }
    }
}

extern "C" void kernel_launch(void* const* d_in, const int* in_sizes, int n_in,
                              void* d_out, int out_size, void* d_ws, size_t ws_size,
                              hipStream_t stream) {
    const float* x = (const float*)d_in[0];
    float* out = (float*)d_out;

    const int planes = in_sizes[0] / (IMG_H * IMG_W);   // 16*3 = 48
    const size_t planeElems = (size_t)planes * IMG_H * IMG_W;

    float* buf0 = (float*)d_ws;
    float* buf1 = buf0 + planeElems;

    dim3 grid(IMG_W / TILE, IMG_H / TILE, planes);
    dim3 block(256);

    const float* src = x;
    for (int it = 0; it < 25; ++it) {
        float* dst = (it & 1) ? buf1 : buf0;
        satpyr_iter<<<grid, block, 0, stream>>>(src, dst, out, it > 0 ? 1 : 0);
        src = dst;
    }
}